// GCNConvLayer_75677323756038
// MI455X (gfx1250) — compile-verified
//
#include <hip/hip_runtime.h>
#include <hip/hip_bf16.h>

#define N_NODES 50000
#define N_EDGES 800000
#define DIM 128
#define BN_EPS 1e-5f

typedef __attribute__((ext_vector_type(2))) float v2f;
typedef __attribute__((ext_vector_type(8))) float v8f;

// ---------------- workspace layout (floats) ----------------
// [0,               N*D)        xw   = x @ W^T
// [N*D,             2*N*D)      h    = aggregated messages
// [2*N*D,           2*N*D+N)    deg  -> overwritten in-place with deg^{-1/2}
// [2*N*D+N, +128)               sum
// [.. +128)                     sumsq
// [.. +128)                     scaleA  (gamma * rstd)
// [.. +128)                     shiftC  (beta - mean * scaleA)

// ---- degree: deg[i] = 1 (self loop) + #edges with col == i ----
__global__ void deg_init_kernel(float* __restrict__ deg) {
    int i = blockIdx.x * blockDim.x + threadIdx.x;
    if (i < N_NODES) deg[i] = 1.0f;
}

__global__ void deg_accum_kernel(const int* __restrict__ col, float* __restrict__ deg) {
    int e = blockIdx.x * blockDim.x + threadIdx.x;
    if (e < N_EDGES) atomicAdd(&deg[col[e]], 1.0f);
}

__global__ void dinv_kernel(float* __restrict__ deg) {
    int i = blockIdx.x * blockDim.x + threadIdx.x;
    if (i < N_NODES) deg[i] = rsqrtf(deg[i]);  // deg >= 1 always (self loops)
}

// ---- GEMM: xw[n][c] = sum_k x[n][k] * W[c][k], via V_WMMA_F32_16X16X4_F32 ----
// Block: 256 threads = 8 waves. Wave w computes the 16x16 tile
// rows [blockIdx.x*16, +16) x cols [w*16, +16). K loop in steps of 4.
__global__ void __launch_bounds__(256)
gemm_wmma_kernel(const float* __restrict__ x, const float* __restrict__ W,
                 float* __restrict__ xw) {
    const int wave = threadIdx.x >> 5;
    const int lane = threadIdx.x & 31;
    const int half = lane >> 4;   // 0: lanes 0-15, 1: lanes 16-31
    const int l16  = lane & 15;
    const int rowBase = blockIdx.x << 4;   // 3125 blocks * 16 = 50000 exactly
    const int colBase = wave << 4;         // 8 waves * 16 = 128 cols

    // A 16x4 f32: lane l16 holds row rowBase+l16; VGPR pair = K {2*half, 2*half+1}
    const float* arow = x + (size_t)(rowBase + l16) * DIM + 2 * half;
    // B 4x16 f32: lane l16 holds col colBase+l16; B[k][n] = W[n][k] contiguous in k
    const float* brow = W + (size_t)(colBase + l16) * DIM + 2 * half;

    v8f c = {};
#pragma unroll
    for (int k = 0; k < DIM; k += 4) {
        v2f a = *(const v2f*)(arow + k);
        v2f b = *(const v2f*)(brow + k);
        c = __builtin_amdgcn_wmma_f32_16x16x4_f32(
            /*neg_a=*/false, a, /*neg_b=*/false, b,
            /*c_mod=*/(short)0, c, /*reuse_a=*/false, /*reuse_b=*/false);
    }

    // C/D layout: VGPR r -> row (r + 8*half), col l16
#pragma unroll
    for (int r = 0; r < 8; ++r) {
        xw[(size_t)(rowBase + r + 8 * half) * DIM + colBase + l16] = c[r];
    }
}

// ---- h init: self-loop message h[i] = xw[i] * dinv[i]^2; also fully rewrites h ----
__global__ void h_init_kernel(const float* __restrict__ xw, const float* __restrict__ dinv,
                              float* __restrict__ h) {
    int idx = blockIdx.x * blockDim.x + threadIdx.x;  // over N*D
    int i = idx >> 7;  // / DIM
    float di = dinv[i];
    h[idx] = xw[idx] * di * di;
}

__global__ void zero_stats_kernel(float* __restrict__ stats) {
    int t = threadIdx.x;
    if (t < 2 * DIM) stats[t] = 0.0f;  // sum + sumsq
}

// ---- edge scatter: h[dst] += xw[src] * dinv[src]*dinv[dst]; one wave per edge ----
__global__ void __launch_bounds__(256)
scatter_kernel(const int* __restrict__ rows, const int* __restrict__ cols,
               const float* __restrict__ dinv, const float* __restrict__ xw,
               float* __restrict__ h) {
    int gid  = blockIdx.x * blockDim.x + threadIdx.x;
    int e    = gid >> 5;
    int lane = gid & 31;
    if (e >= N_EDGES) return;
    int src = rows[e];
    int dst = cols[e];
    float nrm = dinv[src] * dinv[dst];
    float4 v = *((const float4*)(xw + (size_t)src * DIM) + lane);
    float* hp = h + (size_t)dst * DIM + lane * 4;
    atomicAdd(hp + 0, v.x * nrm);
    atomicAdd(hp + 1, v.y * nrm);
    atomicAdd(hp + 2, v.z * nrm);
    atomicAdd(hp + 3, v.w * nrm);
}

// ---- BN stats: per-dim sum / sumsq, coalesced (thread d reads column d) ----
__global__ void __launch_bounds__(128)
stats_kernel(const float* __restrict__ h, float* __restrict__ sum,
             float* __restrict__ sumsq) {
    int d = threadIdx.x;  // blockDim.x == 128
    int rowsPerBlock = (N_NODES + gridDim.x - 1) / gridDim.x;
    int r0 = blockIdx.x * rowsPerBlock;
    int r1 = min(r0 + rowsPerBlock, N_NODES);
    float s = 0.0f, s2 = 0.0f;
    for (int r = r0; r < r1; ++r) {
        float v = h[(size_t)r * DIM + d];
        s += v;
        s2 += v * v;
    }
    atomicAdd(&sum[d], s);
    atomicAdd(&sumsq[d], s2);
}

// ---- finalize stats: scaleA = gamma*rstd, shiftC = beta - mean*scaleA ----
__global__ void finalize_stats_kernel(const float* __restrict__ sum,
                                      const float* __restrict__ sumsq,
                                      const float* __restrict__ gamma,
                                      const float* __restrict__ beta,
                                      float* __restrict__ scaleA,
                                      float* __restrict__ shiftC) {
    int d = threadIdx.x;
    if (d >= DIM) return;
    const float invN = 1.0f / (float)N_NODES;
    float mean = sum[d] * invN;
    float var  = sumsq[d] * invN - mean * mean;
    float rstd = rsqrtf(var + BN_EPS);
    float a = gamma[d] * rstd;
    scaleA[d] = a;
    shiftC[d] = beta[d] - mean * a;
}

// ---- epilogue: out = x + relu(h * scaleA + shiftC)  (BN bias b cancels in BN) ----
__global__ void epilogue_kernel(const float* __restrict__ x, const float* __restrict__ h,
                                const float* __restrict__ scaleA,
                                const float* __restrict__ shiftC,
                                float* __restrict__ out) {
    int idx = blockIdx.x * blockDim.x + threadIdx.x;  // over N*D
    int d = idx & (DIM - 1);
    float v = fmaf(h[idx], scaleA[d], shiftC[d]);
    out[idx] = x[idx] + fmaxf(v, 0.0f);
}

extern "C" void kernel_launch(void* const* d_in, const int* in_sizes, int n_in,
                              void* d_out, int out_size, void* d_ws, size_t ws_size,
                              hipStream_t stream) {
    const float* x      = (const float*)d_in[0];
    const int*   eidx   = (const int*)d_in[1];   // [2, E] flat: rows then cols
    const float* W      = (const float*)d_in[2];
    // d_in[3] = b (cancels exactly through batch-norm; unused)
    const float* gamma  = (const float*)d_in[4];
    const float* beta   = (const float*)d_in[5];
    float*       out    = (float*)d_out;

    const int* rows = eidx;
    const int* cols = eidx + N_EDGES;

    float* ws     = (float*)d_ws;
    float* xw     = ws;                                  // N*D
    float* h      = ws + (size_t)N_NODES * DIM;          // N*D
    float* deg    = ws + (size_t)2 * N_NODES * DIM;      // N  (becomes dinv in place)
    float* sum    = deg + N_NODES;                       // 128
    float* sumsq  = sum + DIM;                           // 128
    float* scaleA = sumsq + DIM;                         // 128
    float* shiftC = scaleA + DIM;                        // 128

    const int T = 256;
    // 1) degrees (with self loops), then deg^{-1/2} in place
    deg_init_kernel<<<(N_NODES + T - 1) / T, T, 0, stream>>>(deg);
    deg_accum_kernel<<<(N_EDGES + T - 1) / T, T, 0, stream>>>(cols, deg);
    dinv_kernel<<<(N_NODES + T - 1) / T, T, 0, stream>>>(deg);
    // 2) xw = x @ W^T via f32 WMMA (50000 rows / 16 = 3125 blocks, 8 waves each)
    gemm_wmma_kernel<<<N_NODES / 16, 256, 0, stream>>>(x, W, xw);
    // 3) h = self-loop term; zero BN accumulators
    h_init_kernel<<<(N_NODES * DIM) / T, T, 0, stream>>>(xw, deg, h);
    zero_stats_kernel<<<1, 256, 0, stream>>>(sum);
    // 4) edge scatter-add (one wave per edge, float4 gather, scalar f32 atomics)
    {
        long long totalThreads = (long long)N_EDGES * 32;
        int blocks = (int)((totalThreads + T - 1) / T);
        scatter_kernel<<<blocks, T, 0, stream>>>(rows, cols, deg, xw, h);
    }
    // 5) batch-norm statistics + epilogue
    stats_kernel<<<250, 128, 0, stream>>>(h, sum, sumsq);
    finalize_stats_kernel<<<1, 128, 0, stream>>>(sum, sumsq, gamma, beta, scaleA, shiftC);
    epilogue_kernel<<<(N_NODES * DIM) / T, T, 0, stream>>>(x, h, scaleA, shiftC, out);
}